// lidar_projection_88527865905722
// MI455X (gfx1250) — compile-verified
//
#include <hip/hip_runtime.h>

// ---------------- static config (mirrors reference) ----------------
constexpr int B_   = 8;
constexpr int N_   = 6;
constexpr int IMH  = 512;
constexpr int IMW  = 1408;
constexpr int DS_  = 16;
constexpr int FH   = IMH / DS_;   // 32
constexpr int FW   = IMW / DS_;   // 88
constexpr int C_   = 64;
constexpr int NX0  = 200;
constexpr int NX1  = 200;
constexpr int NPIX = FH * FW;     // 2816 points per camera image
constexpr int NIMG = B_ * N_;     // 48
constexpr int NP   = NIMG * NPIX; // 135168 total frustum points
constexpr int OUTC = C_ * NX0 * NX1; // per-batch output elements

typedef float v2f __attribute__((ext_vector_type(2)));
typedef float v8f __attribute__((ext_vector_type(8)));

// ---------------- kernel 0: zero the output ----------------
__global__ void zero_f4(float4* __restrict__ out, int n4) {
  int i = blockIdx.x * blockDim.x + threadIdx.x;
  if (i < n4) out[i] = make_float4(0.f, 0.f, 0.f, 0.f);
}

// ---------------- kernel 1: 16x16 max-pool of depth ----------------
// One thread per output pixel; 16 rows x 4 float4 loads. A wave covers a
// contiguous 2 KB span per (r,k) step, every 128B line is fully consumed
// across the unrolled k loop -> single HBM pass over the 138 MB input.
__global__ void __launch_bounds__(256)
depth_maxpool(const float* __restrict__ lidar, float* __restrict__ depth) {
  int p = blockIdx.x * blockDim.x + threadIdx.x;
  if (p >= NP) return;
  int fw  = p % FW;
  int t   = p / FW;
  int fh  = t % FH;
  int img = t / FH;
  const float* base = lidar + (size_t)img * IMH * IMW + (size_t)(fh * DS_) * IMW + fw * DS_;
  float m = -1e30f;
#pragma unroll
  for (int r = 0; r < DS_; ++r) {
    const float4* row = (const float4*)(base + (size_t)r * IMW);
#pragma unroll
    for (int k = 0; k < 4; ++k) {
      float4 v = row[k];
      m = fmaxf(m, fmaxf(fmaxf(v.x, v.y), fmaxf(v.z, v.w)));
    }
  }
  if (m == 0.0f) m = 200.0f;   // zeros -> 200 m, as in reference
  depth[p] = m;
}

// ---------------- 3x3 inverse (adjugate) ----------------
__device__ inline void inv3(const float* __restrict__ m, float* __restrict__ o) {
  float a = m[0], b = m[1], c = m[2];
  float d = m[3], e = m[4], f = m[5];
  float g = m[6], h = m[7], i = m[8];
  float A  = e * i - f * h;
  float Bm = f * g - d * i;
  float Cm = d * h - e * g;
  float r  = 1.0f / (a * A + b * Bm + c * Cm);
  o[0] = A * r;            o[1] = (c * h - b * i) * r;  o[2] = (b * f - c * e) * r;
  o[3] = Bm * r;           o[4] = (a * i - c * g) * r;  o[5] = (c * d - a * f) * r;
  o[6] = Cm * r;           o[7] = (b * g - a * h) * r;  o[8] = (a * e - b * d) * r;
}

// ---------------- kernel 2: geometry via V_WMMA_F32_16X16X4_F32 ----------------
// Each wave transforms 16 homogeneous points by the 3x4 matrix [rots*inv(K) | trans]
// with a single f32 WMMA: A(16x4) = points, B(4x16) = transposed matrix (cols>=3 zero).
// A layout (ISA 7.12.2, 32-bit A 16x4): lanes 0-15 hold K=0,1; lanes 16-31 hold K=2,3.
constexpr int PTS_PER_WAVE   = 16;
constexpr int WAVES_PER_BLK  = 8;
constexpr int PTS_PER_BLK    = PTS_PER_WAVE * WAVES_PER_BLK; // 128
constexpr int BLKS_PER_IMG   = NPIX / PTS_PER_BLK;           // 22 (exact)

__global__ void __launch_bounds__(256)
geom_wmma(const float* __restrict__ rots, const float* __restrict__ trans,
          const float* __restrict__ intrins, const float* __restrict__ post_rots,
          const float* __restrict__ post_trans, const float* __restrict__ depth,
          int* __restrict__ seg) {
  __shared__ float buf[WAVES_PER_BLK][256];

  int tid  = threadIdx.x;
  int wave = tid >> 5;
  int lane = tid & 31;
  int img   = blockIdx.x / BLKS_PER_IMG;
  int chunk = blockIdx.x % BLKS_PER_IMG;
  int pbase = chunk * PTS_PER_BLK + wave * PTS_PER_WAVE;
  int mrow  = lane & 15;          // which of the 16 points this lane describes
  int pimg  = pbase + mrow;       // < NPIX always (exact tiling)

  int fh = pimg / FW, fw = pimg % FW;
  float xs = fw * ((float)(IMW - 1) / (float)(FW - 1));
  float ys = fh * ((float)(IMH - 1) / (float)(FH - 1));

  const float* Rp = rots      + img * 9;
  const float* Kp = intrins   + img * 9;
  const float* Pp = post_rots + img * 9;
  const float* pt = post_trans + img * 3;
  const float* tr = trans      + img * 3;

  float iK[9], iP[9];
  inv3(Kp, iK);
  inv3(Pp, iP);

  // M (3x4) = [ rots * inv(intrins) | trans ]
  float M[12];
#pragma unroll
  for (int i = 0; i < 3; ++i) {
#pragma unroll
    for (int j = 0; j < 3; ++j)
      M[i * 4 + j] = Rp[i * 3 + 0] * iK[0 * 3 + j] +
                     Rp[i * 3 + 1] * iK[1 * 3 + j] +
                     Rp[i * 3 + 2] * iK[2 * 3 + j];
    M[i * 4 + 3] = tr[i];
  }

  float dep = depth[img * NPIX + pimg];
  float fx = xs  - pt[0];
  float fy = ys  - pt[1];
  float fz = 1.f - pt[2];
  float q0 = (iP[0] * fx + iP[1] * fy + iP[2] * fz) * dep;
  float q1 = (iP[3] * fx + iP[4] * fy + iP[5] * fz) * dep;
  float q2 = (iP[6] * fx + iP[7] * fy + iP[8] * fz) * dep;

  // A: point m, K = {q0, q1, q2, 1}
  v2f a;
  if (lane < 16) { a.x = q0; a.y = q1; }
  else           { a.x = q2; a.y = 1.0f; }

  // B[k][n] = M[n][k] for n<3 else 0  (so D[m][n] = ego coordinate n of point m)
  v2f bm;
  int nn = lane & 15;
  if (nn < 3) {
    const float* row = &M[nn * 4];
    if (lane < 16) { bm.x = row[0]; bm.y = row[1]; }
    else           { bm.x = row[2]; bm.y = row[3]; }
  } else {
    bm.x = 0.f; bm.y = 0.f;
  }

  v8f cz = {};
  // D = A(16x4) x B(4x16) + 0
  v8f dd = __builtin_amdgcn_wmma_f32_16x16x4_f32(
      /*neg_a=*/false, a, /*neg_b=*/false, bm,
      /*c_mod=*/(short)0, cz, /*reuse_a=*/false, /*reuse_b=*/false);

  // round-trip D through LDS to gather each point's 3 coords into one lane
#pragma unroll
  for (int r = 0; r < 8; ++r) buf[wave][r * 32 + lane] = dd[r];
  __syncthreads();

  if (lane < 16) {
    int r  = lane & 7;             // D VGPR holding row M=lane
    int lb = (lane >> 3) * 16;     // lane half holding M or M+8
    float X = buf[wave][r * 32 + lb + 0];
    float Y = buf[wave][r * 32 + lb + 1];
    float Z = buf[wave][r * 32 + lb + 2];
    // g = trunc((pts - (BX - DX/2)) / DX); BX-DX/2 = (-50,-50,-10), DX=(0.5,0.5,20)
    int gx = (int)((X + 50.0f) / 0.5f);
    int gy = (int)((Y + 50.0f) / 0.5f);
    int gz = (int)((Z + 10.0f) / 20.0f);
    bool valid = (gx >= 0) & (gx < NX0) & (gy >= 0) & (gy < NX1) & (gz == 0);
    seg[img * NPIX + pbase + lane] = valid ? (gx * NX1 + gy) : -1;
  }
}

// ---------------- kernel 3: scatter-add features into voxel grid ----------------
// Threads map to consecutive pixels -> feature loads coalesced per channel step.
// Output RMW stays in L2 (82 MB grid << 192 MB L2).
__global__ void __launch_bounds__(256)
scatter_add(const float* __restrict__ x, const int* __restrict__ seg,
            float* __restrict__ out) {
  int p = blockIdx.x * blockDim.x + threadIdx.x;   // global point, < NP (exact grid)
  int img = p / NPIX;
  int pix = p % NPIX;
  int s = seg[p];
  if (s < 0) return;
  int b = img / N_;
  const float* xb = x + (size_t)img * C_ * NPIX + pix;   // +c*NPIX per channel
  float* ob = out + (size_t)b * OUTC + s;                // +c*NX0*NX1 per channel
#pragma unroll 4
  for (int c = 0; c < C_; ++c) {
    unsafeAtomicAdd(ob + c * (NX0 * NX1), xb[(size_t)c * NPIX]);
  }
}

// ---------------- launch ----------------
extern "C" void kernel_launch(void* const* d_in, const int* in_sizes, int n_in,
                              void* d_out, int out_size, void* d_ws, size_t ws_size,
                              hipStream_t stream) {
  const float* x          = (const float*)d_in[0];
  const float* lidar      = (const float*)d_in[1];
  const float* rots       = (const float*)d_in[2];
  const float* trans      = (const float*)d_in[3];
  const float* intrins    = (const float*)d_in[4];
  const float* post_rots  = (const float*)d_in[5];
  const float* post_trans = (const float*)d_in[6];
  float* out = (float*)d_out;

  float* depth_ws = (float*)d_ws;          // NP floats
  int*   seg_ws   = (int*)d_ws + NP;       // NP ints

  int n4 = out_size / 4;                   // out_size = 20,480,000 (div by 4)
  zero_f4<<<(n4 + 255) / 256, 256, 0, stream>>>((float4*)out, n4);

  depth_maxpool<<<(NP + 255) / 256, 256, 0, stream>>>(lidar, depth_ws);

  geom_wmma<<<NIMG * BLKS_PER_IMG, 256, 0, stream>>>(
      rots, trans, intrins, post_rots, post_trans, depth_ws, seg_ws);

  scatter_add<<<NP / 256, 256, 0, stream>>>(x, seg_ws, out);
}